// Large_batch_queue_67138928771106
// MI455X (gfx1250) — compile-verified
//
#include <hip/hip_runtime.h>
#include <stdint.h>

constexpr int kNProposals = 1024;
constexpr int kFeat       = 256;
constexpr int kQueue      = 100000;
constexpr int kF4PerRow   = kFeat / 4;   // 64 float4 per row

// ---------------------------------------------------------------------------
// Kernel 1: sorted unique of 1024 labels (one block, O(n^2) with b128 LDS reads)
// Writes: uniq[r] = r-th smallest distinct label (r < U), *ucount = U
// ---------------------------------------------------------------------------
__global__ __launch_bounds__(1024)
void uniq_kernel(const int* __restrict__ labels,
                 int* __restrict__ uniq,
                 int* __restrict__ ucount) {
    __shared__ __align__(16) int vals[kNProposals];
    __shared__ __align__(16) int keys[kNProposals];   // (val<<1) | isFirstOccurrence

    const int t = threadIdx.x;
    const int x = labels[t];
    vals[t] = x;
    __syncthreads();

    // Phase 1: am I the first occurrence of x? (check j < t)
    int isRep = 1;
    {
        const int4* v4 = (const int4*)vals;
        const int nfull = t >> 2;
        for (int c = 0; c < nfull; ++c) {
            int4 v = v4[c];
            isRep &= (v.x != x) & (v.y != x) & (v.z != x) & (v.w != x);
        }
        for (int j = nfull << 2; j < t; ++j) isRep &= (vals[j] != x);
    }
    keys[t] = (x << 1) | isRep;
    __syncthreads();

    // Phase 2: rank = # distinct values < x
    int rank = 0;
    {
        const int4* k4 = (const int4*)keys;
        for (int c = 0; c < kNProposals / 4; ++c) {
            int4 k = k4[c];
            rank += (k.x & 1) & (int)((k.x >> 1) < x);
            rank += (k.y & 1) & (int)((k.y >> 1) < x);
            rank += (k.z & 1) & (int)((k.z >> 1) < x);
            rank += (k.w & 1) & (int)((k.w >> 1) < x);
        }
    }
    if (isRep) uniq[rank] = x;

    // Phase 3: U = sum of rep flags (tree reduce in LDS)
    __syncthreads();
    vals[t] = isRep;
    __syncthreads();
    for (int s = 512; s > 0; s >>= 1) {
        if (t < s) vals[t] += vals[t + s];
        __syncthreads();
    }
    if (t == 0) *ucount = vals[0];
}

// ---------------------------------------------------------------------------
// Kernel 2: bulk queue copy via CDNA5 async global<->LDS data mover.
// out_queue[row] = (row < U) ? features[row] : queue_in[row], 16B per lane.
// Grid: 100000*64 float4 = 6,400,000 lanes = 25000 blocks x 256 (exact).
// ---------------------------------------------------------------------------
__global__ __launch_bounds__(256)
void queue_copy_async(const float4* __restrict__ features,
                      const float4* __restrict__ qin,
                      float4* __restrict__ qout,
                      const int* __restrict__ ucount) {
    __shared__ __align__(16) float4 stage[256];

    const int U = *ucount;
    const long long idx = (long long)blockIdx.x * 256 + threadIdx.x;  // float4 index
    const int row = (int)(idx >> 6);                                   // /64

    const float4* src = (row < U) ? features : qin;  // features row r aliases offset r*64

    const unsigned long long gsrc = (unsigned long long)(uintptr_t)(src + idx);
    const unsigned long long gdst = (unsigned long long)(uintptr_t)(qout + idx);
    // AS3 byte offset of this thread's staging slot (low 32 bits of generic addr)
    const unsigned ldsOff = (unsigned)(uintptr_t)(&stage[threadIdx.x]);

    // Async DMA: global -> LDS (16B per lane), tracked by ASYNCcnt
    asm volatile("global_load_async_to_lds_b128 %0, %1, off"
                 :: "v"(ldsOff), "v"(gsrc) : "memory");
    // RAW on LDS: wait for the async load to land before the async store reads it
    asm volatile("s_wait_asynccnt 0x0" ::: "memory");
    // Async DMA: LDS -> global; completion covered by implicit wait-idle at endpgm
    asm volatile("global_store_async_from_lds_b128 %0, %1, off"
                 :: "v"(gdst), "v"(ldsOff) : "memory");
}

// ---------------------------------------------------------------------------
// Kernel 3: label region. out_label[i] = (i < U) ? float(uniq[i]) : qlabel_in[i]
// ---------------------------------------------------------------------------
__global__ __launch_bounds__(256)
void label_copy(const float* __restrict__ qlabel_in,
                const int* __restrict__ uniq,
                const int* __restrict__ ucount,
                float* __restrict__ out_label) {
    const int i = blockIdx.x * 256 + threadIdx.x;
    if (i >= kQueue) return;
    const int U = *ucount;
    out_label[i] = (i < U) ? (float)uniq[i] : qlabel_in[i];
}

extern "C" void kernel_launch(void* const* d_in, const int* in_sizes, int n_in,
                              void* d_out, int out_size, void* d_ws, size_t ws_size,
                              hipStream_t stream) {
    const float* features  = (const float*)d_in[0];   // (1024, 256) f32
    const int*   labels    = (const int*)  d_in[1];   // (1024,) i32
    const float* qin       = (const float*)d_in[2];   // (100000, 256) f32
    const float* qlabel_in = (const float*)d_in[3];   // (100000,) f32

    float* out_queue = (float*)d_out;                                  // 25,600,000
    float* out_label = (float*)d_out + (size_t)kQueue * kFeat;         // +100,000

    int* uniq   = (int*)d_ws;           // 1024 ints
    int* ucount = uniq + kNProposals;   // 1 int

    uniq_kernel<<<1, 1024, 0, stream>>>(labels, uniq, ucount);

    const int nF4 = kQueue * kF4PerRow;               // 6,400,000 (exact multiple of 256)
    queue_copy_async<<<nF4 / 256, 256, 0, stream>>>(
        (const float4*)features, (const float4*)qin, (float4*)out_queue, ucount);

    label_copy<<<(kQueue + 255) / 256, 256, 0, stream>>>(qlabel_in, uniq, ucount, out_label);
}